// GraphAttention_76914274337237
// MI455X (gfx1250) — compile-verified
//
#include <hip/hip_runtime.h>
#include <hip/hip_bf16.h>

typedef __attribute__((ext_vector_type(2))) float v2f;
typedef __attribute__((ext_vector_type(8))) float v8f;

#define D 128
#define NEG_SLOPE 0.01f

// Relaxed agent-scope FP32 atomic add -> global_atomic_add_f32 in the L2
// atomic units (no system scope, no ordering cost, no CAS expansion).
__device__ __forceinline__ void atomic_add_f32(float* p, float v) {
    __hip_atomic_fetch_add(p, v, __ATOMIC_RELAXED, __HIP_MEMORY_SCOPE_AGENT);
}

// ---------------------------------------------------------------------------
// Kernel 1: Wh = x @ W^T using V_WMMA_F32_16X16X4_F32 (full fp32 precision).
// Block = 256 threads = 8 wave32s. Block handles 16 rows of x; wave w computes
// the 16x16 output tile for columns [16w, 16w+16). W (64KB) is staged
// transposed in LDS so B-fragment reads are contiguous across lanes.
// ---------------------------------------------------------------------------
__global__ __launch_bounds__(256) void wh_gemm_wmma(
    const float* __restrict__ x, const float* __restrict__ W,
    float* __restrict__ Wh, int N)
{
    __shared__ float Wt[D * D];   // Wt[k*D + j] = W[j*D + k]  (= B[k][j])

    const int tid = threadIdx.x;
    for (int idx = tid; idx < D * D; idx += 256) {
        const int j = idx >> 7;
        const int k = idx & (D - 1);
        Wt[k * D + j] = W[idx];
    }
    __syncthreads();

    const int wave    = tid >> 5;
    const int lane    = tid & 31;
    const int half    = lane >> 4;      // 0: lanes 0-15, 1: lanes 16-31
    const int l16     = lane & 15;
    const int rowBase = blockIdx.x * 16;
    const int colBase = wave * 16;

    // A-matrix row owned by this lane (M = l16 in both lane halves)
    int arow = rowBase + l16;
    if (arow >= N) arow = N - 1;                 // clamp (tail safety)
    const float* __restrict__ xrow = x + (size_t)arow * D;

    v8f c = {0.f, 0.f, 0.f, 0.f, 0.f, 0.f, 0.f, 0.f};

#pragma unroll
    for (int k0 = 0; k0 < D; k0 += 4) {
        // A 16x4 frag (ISA 7.12.2): lanes 0-15 -> K=k0,k0+1 ; lanes 16-31 -> K=k0+2,k0+3
        v2f a;
        a.x = xrow[k0 + 2 * half + 0];
        a.y = xrow[k0 + 2 * half + 1];
        // B 4x16 frag: B[k][n] = Wt[k*D + colBase + n], same K split across halves
        v2f b;
        b.x = Wt[(k0 + 2 * half + 0) * D + colBase + l16];
        b.y = Wt[(k0 + 2 * half + 1) * D + colBase + l16];
        c = __builtin_amdgcn_wmma_f32_16x16x4_f32(
                /*neg_a=*/false, a, /*neg_b=*/false, b,
                /*c_mod=*/(short)0, c, /*reuse_a=*/false, /*reuse_b=*/false);
    }

    // C/D layout: VGPR r -> M = r (lanes 0-15) / M = r+8 (lanes 16-31), N = l16
#pragma unroll
    for (int r = 0; r < 8; ++r) {
        const int row = rowBase + r + 8 * half;
        if (row < N) Wh[(size_t)row * D + colBase + l16] = c[r];
    }
}

// ---------------------------------------------------------------------------
// Kernel 2: s_src[i] = Wh[i,:] . a[0:128], s_dst[i] = Wh[i,:] . a[128:256]
// One wave32 per row; shfl_xor tree reduction (wave32-correct).
// ---------------------------------------------------------------------------
__global__ __launch_bounds__(256) void score_kernel(
    const float* __restrict__ Wh, const float* __restrict__ a,
    float* __restrict__ s_src, float* __restrict__ s_dst, int N)
{
    const int wave = threadIdx.x >> 5;
    const int lane = threadIdx.x & 31;
    const int row  = blockIdx.x * 8 + wave;
    if (row >= N) return;

    const float* __restrict__ wr = Wh + (size_t)row * D;
    float ps = 0.f, pd = 0.f;
#pragma unroll
    for (int k = 0; k < 4; ++k) {
        const float v = wr[lane + 32 * k];
        ps += v * a[lane + 32 * k];
        pd += v * a[D + lane + 32 * k];
    }
#pragma unroll
    for (int off = 16; off > 0; off >>= 1) {
        ps += __shfl_xor(ps, off, 32);
        pd += __shfl_xor(pd, off, 32);
    }
    if (lane == 0) { s_src[row] = ps; s_dst[row] = pd; }
}

// ---------------------------------------------------------------------------
// Kernel 3: per-edge attention logits -> num[t]; denom via scalar FP atomics.
// Edges t<E come from edge_index; t>=E are self loops.
// ---------------------------------------------------------------------------
__global__ void edge_kernel(
    const int* __restrict__ ei,
    const float* __restrict__ s_src, const float* __restrict__ s_dst,
    float* __restrict__ num, float* __restrict__ denom, int E, int T)
{
    const int t = blockIdx.x * blockDim.x + threadIdx.x;
    if (t >= T) return;
    const int src = (t < E) ? ei[t]     : (t - E);
    const int dst = (t < E) ? ei[E + t] : (t - E);
    float e = s_src[src] + s_dst[dst];
    e = (e > 0.f) ? e : NEG_SLOPE * e;
    const float n = expf(e);
    num[t] = n;
    atomic_add_f32(&denom[dst], n);
}

// ---------------------------------------------------------------------------
// Kernel 4: agg[dst,:] += num[t] * Wh[src,:].  One wave32 per edge; float4
// gather + 4 hardware FP atomics per lane. Wh (51MB) and agg (51MB) both fit
// in the 192MB L2, and the L2 owns the atomic units -> near-L2-bandwidth RMW.
// ---------------------------------------------------------------------------
__global__ __launch_bounds__(256) void agg_scatter(
    const int* __restrict__ ei, const float* __restrict__ num,
    const float* __restrict__ Wh, float* __restrict__ agg, int E, int T)
{
    const int wave = threadIdx.x >> 5;
    const int lane = threadIdx.x & 31;
    const int t    = blockIdx.x * 8 + wave;
    if (t >= T) return;
    const int src = (t < E) ? ei[t]     : (t - E);
    const int dst = (t < E) ? ei[E + t] : (t - E);
    const float n = num[t];

    const float4 v = ((const float4*)(Wh + (size_t)src * D))[lane];
    float* ad = agg + (size_t)dst * D + lane * 4;
    atomic_add_f32(ad + 0, n * v.x);
    atomic_add_f32(ad + 1, n * v.y);
    atomic_add_f32(ad + 2, n * v.z);
    atomic_add_f32(ad + 3, n * v.w);
}

// ---------------------------------------------------------------------------
// Kernel 5: clamp denom (idempotent, race-safe) and normalize agg.
// ---------------------------------------------------------------------------
__global__ __launch_bounds__(128) void normalize_kernel(
    float* __restrict__ agg, float* __restrict__ denom, int N)
{
    const int i = blockIdx.x;
    const int j = threadIdx.x;
    if (i >= N) return;
    const float d = fmaxf(denom[i], 1e-12f);
    agg[(size_t)i * D + j] /= d;
    if (j == 0) denom[i] = d;     // fmax is idempotent -> safe w/o sync
}

__global__ void zero_kernel(float* __restrict__ p, long n)
{
    const long i = (long)blockIdx.x * blockDim.x + threadIdx.x;
    if (i < n) p[i] = 0.f;
}

// ---------------------------------------------------------------------------
extern "C" void kernel_launch(void* const* d_in, const int* in_sizes, int n_in,
                              void* d_out, int out_size, void* d_ws, size_t ws_size,
                              hipStream_t stream)
{
    const float* x  = (const float*)d_in[0];   // (N,128) f32
    const int*   ei = (const int*)  d_in[1];   // (2,E) i32, row-major
    const float* W  = (const float*)d_in[2];   // (128,128) f32
    const float* a  = (const float*)d_in[3];   // (256,) f32

    const int N = in_sizes[0] / D;
    const int E = in_sizes[1] / 2;
    const int T = E + N;

    // d_out layout (return order): agg(N*128) | num(T) | denom(N) | Wh(N*128)
    float* out   = (float*)d_out;
    float* agg   = out;
    float* num   = agg + (size_t)N * D;
    float* denom = num + T;
    float* Wh    = denom + N;

    // workspace: s_src(N) | s_dst(N)
    float* s_src = (float*)d_ws;
    float* s_dst = s_src + N;

    // accumulators must start at zero every call
    const long nAgg = (long)N * D;
    zero_kernel<<<(int)((nAgg + 255) / 256), 256, 0, stream>>>(agg, nAgg);
    zero_kernel<<<(N + 255) / 256, 256, 0, stream>>>(denom, N);

    const int rowTiles = (N + 15) / 16;
    wh_gemm_wmma<<<rowTiles, 256, 0, stream>>>(x, W, Wh, N);
    score_kernel<<<(N + 7) / 8, 256, 0, stream>>>(Wh, a, s_src, s_dst, N);
    edge_kernel<<<(T + 255) / 256, 256, 0, stream>>>(ei, s_src, s_dst, num, denom, E, T);
    agg_scatter<<<(T + 7) / 8, 256, 0, stream>>>(ei, num, Wh, agg, E, T);
    normalize_kernel<<<N, 128, 0, stream>>>(agg, denom, N);
}